// Group_Quantization_90718299226528
// MI455X (gfx1250) — compile-verified
//
#include <hip/hip_runtime.h>

typedef __attribute__((ext_vector_type(2))) float v2f;
typedef __attribute__((ext_vector_type(8))) float v8f;

#define NG 3
#define KD 40
#define NS 1024
#define NB 32
#define TPQ 4096
#define MTOT (NB * TPQ)        /* 131072 vectors per group */
#define CT 491520              /* 30*16384 = flat stride per batch item */
#define S_CHUNK 256            /* codewords staged in LDS per pass */
#define NCHUNK (NS / S_CHUNK)
#define WAVES 8
#define ROWS_PER_BLOCK (WAVES * 16)   /* 128 M-rows per block */
#define Q_ELEMS 15728640

__launch_bounds__(256)
__global__ void vq_argmin_wmma(const float* __restrict__ z,
                               const float* __restrict__ cbs,
                               float* __restrict__ out,
                               float* __restrict__ lossWs) {
    __shared__ float cbuf[S_CHUNK * KD];   // 40 KB codebook chunk
    __shared__ float e2s[NS];              // 4 KB  |e|^2 table
    __shared__ int   bidx[WAVES][16];      // winning index per row
    __shared__ float lred[256];

    const int g    = blockIdx.y;
    const float* cb = cbs + g * NS * KD;
    const int tid  = threadIdx.x;
    const int wave = tid >> 5;
    const int lane = tid & 31;
    const int half = lane >> 4;            // 0: K%4 in {0,1}, 1: {2,3}
    const int l15  = lane & 15;

    // ---- precompute |e|^2 for all 1024 codewords of this group ----
    for (int i = tid; i < NS; i += 256) {
        const float* r = cb + i * KD;
        float s = 0.f;
        #pragma unroll
        for (int k = 0; k < KD; ++k) s += r[k] * r[k];
        e2s[i] = s;
    }

    // ---- load A tile (16 rows x 40 k) into registers, WMMA f32 16x16x4 layout ----
    const int mbase = (blockIdx.x * WAVES + wave) * 16;
    const int m  = mbase + l15;
    const int n  = m >> 12;
    const int t  = m & 4095;
    const float* zrow = z + (size_t)n * CT + (size_t)(g * KD) * TPQ + t;
    v2f a[10];
    #pragma unroll
    for (int k0 = 0; k0 < 10; ++k0) {
        const int k = 4 * k0 + 2 * half;
        a[k0].x = zrow[(size_t)k * TPQ];
        a[k0].y = zrow[(size_t)(k + 1) * TPQ];
    }

    float best[8];
    int   bi[8];
    #pragma unroll
    for (int j = 0; j < 8; ++j) { best[j] = 3.4e38f; bi[j] = 0; }

    __syncthreads();

    // ---- codebook search: 4 chunks x (8 double s-tiles), 20 WMMAs each ----
    for (int ch = 0; ch < NCHUNK; ++ch) {
        const float4* src = (const float4*)(cb + ch * S_CHUNK * KD);
        float4* dst = (float4*)cbuf;
        for (int i = tid; i < (S_CHUNK * KD) / 4; i += 256) dst[i] = src[i];
        __syncthreads();

        #pragma unroll 1
        for (int st = 0; st < S_CHUNK / 16; st += 2) {
            const int sLoc0  = st * 16 + l15;
            const int sGlob0 = ch * S_CHUNK + sLoc0;
            const float* brow0 = cbuf + sLoc0 * KD + 2 * half;
            const float* brow1 = brow0 + 16 * KD;
            v8f c0 = {0.f, 0.f, 0.f, 0.f, 0.f, 0.f, 0.f, 0.f};
            v8f c1 = {0.f, 0.f, 0.f, 0.f, 0.f, 0.f, 0.f, 0.f};
            #pragma unroll
            for (int k0 = 0; k0 < 10; ++k0) {
                v2f b0 = *(const v2f*)(brow0 + 4 * k0);   // 8B aligned
                v2f b1 = *(const v2f*)(brow1 + 4 * k0);
                c0 = __builtin_amdgcn_wmma_f32_16x16x4_f32(
                         false, a[k0], false, b0, (short)0, c0, false, false);
                c1 = __builtin_amdgcn_wmma_f32_16x16x4_f32(
                         false, a[k0], false, b1, (short)0, c1, false, false);
            }
            const float e20 = e2s[sGlob0];
            const float e21 = e2s[sGlob0 + 16];
            // lower s-tile first, strict < : keeps lowest index on ties
            #pragma unroll
            for (int j = 0; j < 8; ++j) {
                const float s0 = fmaf(-2.f, c0[j], e20);  // x^2 term constant per row
                if (s0 < best[j]) { best[j] = s0; bi[j] = sGlob0; }
            }
            #pragma unroll
            for (int j = 0; j < 8; ++j) {
                const float s1 = fmaf(-2.f, c1[j], e21);
                if (s1 < best[j]) { best[j] = s1; bi[j] = sGlob0 + 16; }
            }
        }
        __syncthreads();
    }

    // ---- cross-lane argmin within each 16-lane half (wave32 butterfly) ----
    #pragma unroll
    for (int j = 0; j < 8; ++j) {
        #pragma unroll
        for (int mask = 1; mask < 16; mask <<= 1) {
            const float ob = __shfl_xor(best[j], mask, 32);
            const int   oi = __shfl_xor(bi[j],   mask, 32);
            if (ob < best[j] || (ob == best[j] && oi < bi[j])) {
                best[j] = ob; bi[j] = oi;
            }
        }
    }
    if (l15 == 0) {
        #pragma unroll
        for (int j = 0; j < 8; ++j) bidx[wave][half * 8 + j] = bi[j];
    }
    __syncthreads();

    // ---- writeback: out = x + (q - x); accumulate loss for last group ----
    const int idx = bidx[wave][l15];
    const int mr  = mbase + l15;
    const int nr  = mr >> 12;
    const int tr  = mr & 4095;
    const float* qrow = cb + idx * KD;
    const size_t obase = (size_t)nr * CT + (size_t)(g * KD) * TPQ + tr;
    float acc = 0.f;
    #pragma unroll
    for (int kk = 0; kk < 20; ++kk) {
        const int k = half * 20 + kk;
        const size_t o = obase + (size_t)k * TPQ;
        const float qv = qrow[k];
        const float xv = z[o];
        const float d  = qv - xv;
        out[o] = xv + d;          // exact straight-through arithmetic
        acc += d * d;
    }

    if (g == 2) {   // deterministic per-block partial of commitment loss
        lred[tid] = acc;
        __syncthreads();
        for (int off = 128; off > 0; off >>= 1) {
            if (tid < off) lred[tid] += lred[tid + off];
            __syncthreads();
        }
        if (tid == 0) lossWs[blockIdx.x] = lred[0];
    }
}

__launch_bounds__(256)
__global__ void loss_reduce(const float* __restrict__ ws, float* __restrict__ lossOut) {
    __shared__ float red[256];
    float s = 0.f;
    for (int i = threadIdx.x; i < MTOT / ROWS_PER_BLOCK; i += 256) s += ws[i];
    red[threadIdx.x] = s;
    __syncthreads();
    for (int off = 128; off > 0; off >>= 1) {
        if (threadIdx.x < off) red[threadIdx.x] += red[threadIdx.x + off];
        __syncthreads();
    }
    if (threadIdx.x == 0)
        lossOut[0] = 0.25f * red[0] / ((float)MTOT * (float)KD);
}

extern "C" void kernel_launch(void* const* d_in, const int* in_sizes, int n_in,
                              void* d_out, int out_size, void* d_ws, size_t ws_size,
                              hipStream_t stream) {
    const float* z   = (const float*)d_in[0];
    const float* cbs = (const float*)d_in[1];
    float* out = (float*)d_out;
    float* ws  = (float*)d_ws;

    dim3 grid(MTOT / ROWS_PER_BLOCK, NG);   // 1024 x 3 blocks
    vq_argmin_wmma<<<grid, 256, 0, stream>>>(z, cbs, out, ws);
    loss_reduce<<<1, 256, 0, stream>>>(ws, out + (size_t)Q_ELEMS);
}